// JointLoss_90821378441753
// MI455X (gfx1250) — compile-verified
//
#include <hip/hip_runtime.h>
#include <hip/hip_bf16.h>
#include <math.h>

// Problem constants (SHAPE = (C=2, H=128, W=128, T=16, B=2), KERNEL=(3,3))
#define C_  2
#define H_  128
#define W_  128
#define T_  16
#define B_  2
#define HO  126
#define WO  126
#define NMAT  (HO*WO)        // 15876 casorati matrices (36x16 each)
#define NSITE (C_*H_*W_)     // 32768 sites of 32 contiguous floats (T*B)
#define NSTRIP 16            // strips of 8 w0-positions
#define NCASBLK (HO*NSTRIP)  // 2016 casorati blocks

typedef __attribute__((ext_vector_type(2))) float v2f;
typedef __attribute__((ext_vector_type(8))) float v8f;

// cos/sin(2*pi*n/16)
__device__ const float COS16[16] = {
  1.f, 0.92387953f, 0.70710678f, 0.38268343f, 0.f, -0.38268343f, -0.70710678f, -0.92387953f,
 -1.f,-0.92387953f,-0.70710678f,-0.38268343f, 0.f,  0.38268343f,  0.70710678f,  0.92387953f};
__device__ const float SIN16[16] = {
  0.f, 0.38268343f, 0.70710678f, 0.92387953f, 1.f,  0.92387953f,  0.70710678f,  0.38268343f,
  0.f,-0.38268343f,-0.70710678f,-0.92387953f,-1.f, -0.92387953f, -0.70710678f, -0.38268343f};

// ---------------------------------------------------------------------------
// Kernel 1: per-site (32 contiguous floats) reductions (bandwidth-bound):
//   q0: S_ref, q1: hank_num (DFT nuclear norm of anti-circulant), q2: l1_ref,
//   q3: l2_ref_sq, q4: num1 (=sum|x0-x1|), q5: num2_sq (=sum (x0-x1)^2/2)
// ---------------------------------------------------------------------------
__global__ __launch_bounds__(256) void site_reduce_kernel(
    const float* __restrict__ x, const float* __restrict__ xr, float* __restrict__ ws) {
  const int site = blockIdx.x * 256 + threadIdx.x;   // exactly NSITE threads
  float xb[32], rb[32];
  const float4* xv = reinterpret_cast<const float4*>(x)  + (size_t)site * 8;
  const float4* rv = reinterpret_cast<const float4*>(xr) + (size_t)site * 8;
#pragma unroll
  for (int i = 0; i < 8; ++i) {
    float4 a = xv[i];
    xb[4*i+0] = a.x; xb[4*i+1] = a.y; xb[4*i+2] = a.z; xb[4*i+3] = a.w;
    float4 b = rv[i];
    rb[4*i+0] = b.x; rb[4*i+1] = b.y; rb[4*i+2] = b.z; rb[4*i+3] = b.w;
  }

  float sq = 0.f, l1 = 0.f;
#pragma unroll
  for (int i = 0; i < 32; ++i) { sq = fmaf(rb[i], rb[i], sq); l1 += fabsf(rb[i]); }
  const float Sref = sqrtf(sq);

  float num1 = 0.f, num2 = 0.f;
  float e[16], o[16];
#pragma unroll
  for (int t = 0; t < 16; ++t) {
    const float x0 = xb[2*t], x1 = xb[2*t+1];
    const float d  = x0 - x1;
    num1 += fabsf(d);
    num2 = fmaf(0.5f*d, d, num2);
    e[t] = x0 + x1;   // k2 = 0 of the 2-pt DFT over B
    o[t] = d;         // k2 = 1
  }

  // 16-point real DFT magnitudes; |X[16-k]| == |X[k]| for real input
  float hank = 0.f;
#pragma unroll
  for (int k = 0; k <= 8; ++k) {
    float re_e = 0.f, im_e = 0.f, re_o = 0.f, im_o = 0.f;
#pragma unroll
    for (int t = 0; t < 16; ++t) {
      const float c = COS16[(k*t) & 15], s = SIN16[(k*t) & 15];
      re_e = fmaf(e[t], c, re_e); im_e = fmaf(e[t], s, im_e);
      re_o = fmaf(o[t], c, re_o); im_o = fmaf(o[t], s, im_o);
    }
    const float me = sqrtf(fmaf(re_e, re_e, im_e*im_e));
    const float mo = sqrtf(fmaf(re_o, re_o, im_o*im_o));
    const float wgt = (k == 0 || k == 8) ? 1.f : 2.f;
    hank = fmaf(wgt, me + mo, hank);
  }

  float vals[6] = {Sref, hank, l1, sq, num1, num2};
  __shared__ float sred[8][6];
  const int lane = threadIdx.x & 31, wid = threadIdx.x >> 5;
#pragma unroll
  for (int q = 0; q < 6; ++q) {
    float v = vals[q];
#pragma unroll
    for (int off = 16; off >= 1; off >>= 1) v += __shfl_xor(v, off, 32);
    if (lane == 0) sred[wid][q] = v;
  }
  __syncthreads();
  if (threadIdx.x == 0) {
#pragma unroll
    for (int q = 0; q < 6; ++q) {
      float s = 0.f;
      for (int w = 0; w < 8; ++w) s += sred[w][q];
      ws[blockIdx.x * 6 + q] = s;
    }
  }
}

// ---------------------------------------------------------------------------
// Kernel 2: casorati nuclear norms.
// Block = (h0, strip of 8 w0); async-load the shared (2c x 3h x <=10w x 32)
// tile into LDS once, then one wave per 36x16 matrix:
//   G = M^T M via 9x V_WMMA_F32_16X16X4_F32 (operands fed from LDS),
//   parallel round-robin Jacobi (8 disjoint rotations/round, 15 rounds/sweep),
//   nuclear = sum sqrt(max(eig,0)).
// ---------------------------------------------------------------------------

// Compile-time static part of the LDS tile offset for Casorati row r:
// r = bc*9 + a*3 + bb, bc = b*2 + c; element at tile[(c*3+a)*320 + bb*32 + b + runtime]
__host__ __device__ constexpr int moff(int r) {
  return (((r / 9) & 1) * 3 + (r % 9) / 3) * 320    // (c*3 + a) segment
       + ((r % 9) % 3) * 32                         // bb site offset
       + ((r / 9) >> 1);                            // b element offset
}

__global__ __launch_bounds__(256) void casorati_kernel(
    const float* __restrict__ x, float* __restrict__ ws) {
  __shared__ __align__(16) float tile[6 * 10 * 32];  // 6 (c,a) segments x 10 sites x 32
  __shared__ float Gs[8][16][17];                    // one padded 16x16 Gram per wave
  __shared__ float crot[8][8], srot[8][8];
  __shared__ int   prot[8][8], qrot[8][8];
  __shared__ float wsum[8];

  const int tid  = threadIdx.x;
  const int wid  = tid >> 5;
  const int lane = tid & 31;
  const int t    = lane & 15;           // column index (0..15)
  const int half = lane >> 4;
  const int h0     = blockIdx.x >> 4;   // 0..125
  const int w0base = (blockIdx.x & 15) * 8;
  const int wsites = (W_ - w0base < 10) ? (W_ - w0base) : 10;  // tile width (sites)
  const int nmat   = (WO - w0base < 8)  ? (WO - w0base) : 8;   // matrices in strip

  // ---- async global -> LDS tile load (ASYNCcnt path, no VGPR staging) ----
  const int total16 = 6 * wsites * 8;   // 16-byte units
  for (int i = tid; i < total16; i += 256) {
    const int seg = i / (wsites * 8);
    const int rem = i - seg * (wsites * 8);
    const int c = seg / 3, a = seg - c * 3;
    const size_t goff = ((((size_t)c * H_ + (h0 + a)) * W_ + w0base) << 5) + ((size_t)rem << 2);
    const uint64_t ga = (uint64_t)(uintptr_t)(x + goff);
    const uint32_t la = (uint32_t)(uintptr_t)(&tile[seg * 320 + (rem << 2)]);
    asm volatile("global_load_async_to_lds_b128 %0, %1, off"
                 :: "v"(la), "v"(ga) : "memory");
  }
  asm volatile("s_wait_asynccnt 0x0" ::: "memory");
  __syncthreads();

  float nuc = 0.f;
  if (wid < nmat) {                     // wave-uniform: EXEC all-1s for WMMA
    float (*G)[17] = Gs[wid];

    // ---- Gram via WMMA f32 16x16x4, K = 36 in 9 chunks of 4 ----
    // A = M^T and B = M have identical register contents (lane<16: K rows 0,1;
    // lane>=16: K rows 2,3; col = lane%16) -> one v2f feeds both operands.
    // Static offsets are constexpr-selected by `half` (no runtime div/mod).
    const int rbase = wid * 32 + t * 2;
    v8f acc = {0.f, 0.f, 0.f, 0.f, 0.f, 0.f, 0.f, 0.f};
#pragma unroll
    for (int kc = 0; kc < 9; ++kc) {
      const int oA = half ? moff(4*kc + 2) : moff(4*kc + 0);
      const int oB = half ? moff(4*kc + 3) : moff(4*kc + 1);
      v2f ab;
      ab.x = tile[oA + rbase];
      ab.y = tile[oB + rbase];
      acc = __builtin_amdgcn_wmma_f32_16x16x4_f32(
          false, ab, false, ab, (short)0, acc, false, false);
    }
#pragma unroll
    for (int v = 0; v < 8; ++v) G[v + 8*half][t] = acc[v];   // C/D layout
    __builtin_amdgcn_wave_barrier();

    // ---- parallel round-robin Jacobi: 8 disjoint pairs per round ----
    for (int sweep = 0; sweep < 10; ++sweep) {
      for (int r = 0; r < 15; ++r) {
        if (lane < 8) {                 // angle phase: one pair per lane
          const int k = lane;
          // circle-method pair; bounded operands -> branchless conditional
          // subtract instead of runtime %15
          int a0, b0;
          if (k == 0) {
            a0 = 0;
            b0 = 14 + r; if (b0 >= 15) b0 -= 15; b0 += 1;
          } else {
            a0 = k - 1 + r;  if (a0 >= 15) a0 -= 15; a0 += 1;
            b0 = 14 - k + r; if (b0 >= 15) b0 -= 15; b0 += 1;
          }
          const int p = (a0 < b0) ? a0 : b0;
          const int q = (a0 < b0) ? b0 : a0;
          const float app = G[p][p], aqq = G[q][q], apq = G[p][q];
          float cc = 1.f, sc = 0.f;
          if (fabsf(apq) > 1e-9f * (fabsf(app) + fabsf(aqq)) + 1e-30f) {
            const float th = (aqq - app) / (2.f * apq);
            const float tt = copysignf(1.f, th) / (fabsf(th) + sqrtf(fmaf(th, th, 1.f)));
            cc = 1.f / sqrtf(fmaf(tt, tt, 1.f));
            sc = tt * cc;
          }
          crot[wid][k] = cc; srot[wid][k] = sc;
          prot[wid][k] = p;  qrot[wid][k] = q;
        }
        __builtin_amdgcn_wave_barrier();
#pragma unroll
        for (int i = 0; i < 4; ++i) {   // row phase: disjoint rows across pairs
          const int k = i * 2 + half;
          const int p = prot[wid][k], q = qrot[wid][k];
          const float cc = crot[wid][k], sc = srot[wid][k];
          const float rp = G[p][t], rq = G[q][t];
          G[p][t] = cc * rp - sc * rq;
          G[q][t] = sc * rp + cc * rq;
        }
        __builtin_amdgcn_wave_barrier();
#pragma unroll
        for (int i = 0; i < 4; ++i) {   // col phase: disjoint cols across pairs
          const int k = i * 2 + half;
          const int p = prot[wid][k], q = qrot[wid][k];
          const float cc = crot[wid][k], sc = srot[wid][k];
          const float cp = G[t][p], cq = G[t][q];
          G[t][p] = cc * cp - sc * cq;
          G[t][q] = sc * cp + cc * cq;
        }
        __builtin_amdgcn_wave_barrier();
      }
    }

    float v = (half == 0) ? sqrtf(fmaxf(G[t][t], 0.f)) : 0.f;
#pragma unroll
    for (int off = 16; off >= 1; off >>= 1) v += __shfl_xor(v, off, 32);
    nuc = v;
  }

  if (lane == 0) wsum[wid] = nuc;
  __syncthreads();
  if (tid == 0) {
    float s = 0.f;
    for (int w = 0; w < 8; ++w) s += wsum[w];
    ws[1024 + blockIdx.x] = s;          // NCASBLK casorati partials
  }
}

// ---------------------------------------------------------------------------
// Kernel 3: deterministic finalize (single block).
// loss = 1e-3*(cas_nuc/3)/S_ref + 1e-3*hank_num/(sqrt(32)*S_ref)
//      + 1e-2*num1/l1_ref + 1e-2*sqrt(num2_sq)/sqrt(l2_ref_sq)
// ---------------------------------------------------------------------------
__global__ __launch_bounds__(256) void finalize_kernel(
    const float* __restrict__ ws, float* __restrict__ out) {
  __shared__ float red[256];
  const int tid = threadIdx.x;
  float s[7] = {0.f, 0.f, 0.f, 0.f, 0.f, 0.f, 0.f};
  if (tid < 128) {
#pragma unroll
    for (int q = 0; q < 6; ++q) s[q] = ws[tid * 6 + q];
  }
  for (int i = tid; i < NCASBLK; i += 256) s[6] += ws[1024 + i];

  float tot[7];
#pragma unroll
  for (int q = 0; q < 7; ++q) {
    red[tid] = s[q];
    __syncthreads();
    for (int off = 128; off >= 1; off >>= 1) {
      if (tid < off) red[tid] += red[tid + off];
      __syncthreads();
    }
    tot[q] = red[0];
    __syncthreads();
  }

  if (tid == 0) {
    const float Sref    = tot[0];
    const float hankNum = tot[1];
    const float l1ref   = tot[2];
    const float l2refSq = tot[3];
    const float num1    = tot[4];
    const float num2Sq  = tot[5];
    const float casNuc  = tot[6];
    const float loss =
        0.001f * (casNuc * (1.0f / 3.0f)) / Sref +
        0.001f * hankNum / (5.656854249492381f * Sref) +     // sqrt(32)
        0.01f  * num1 / l1ref +
        0.01f  * sqrtf(num2Sq) / sqrtf(l2refSq);
    out[0] = loss;
  }
}

extern "C" void kernel_launch(void* const* d_in, const int* in_sizes, int n_in,
                              void* d_out, int out_size, void* d_ws, size_t ws_size,
                              hipStream_t stream) {
  const float* x  = (const float*)d_in[0];   // images
  const float* xr = (const float*)d_in[1];   // images_regridded
  float* out = (float*)d_out;
  float* ws  = (float*)d_ws;                 // needs (1024 + 2016) floats = 12160 B

  site_reduce_kernel<<<NSITE / 256, 256, 0, stream>>>(x, xr, ws);   // 128 blocks
  casorati_kernel<<<NCASBLK, 256, 0, stream>>>(x, ws);              // 2016 blocks
  finalize_kernel<<<1, 256, 0, stream>>>(ws, out);
}